// GainesEdgeDetect_16166256902747
// MI455X (gfx1250) — compile-verified
//
#include <hip/hip_runtime.h>

// GainesEdgeDetect, single bit-cycle, sel==0 path:
//   out = FSUAbs_bit(x = inp_Pr_i_j, c = cnt_x)
//   t = clamp(c + 2x - 1, 0, 15);  out = (t < 8) ? (1 - x) : x
// Inputs 1,2,3 (other bit planes) and 5 (cnt_y) are dead in the reference output.
// Memory-bound streaming kernel: 2 NT b128 loads + 1 NT b128 store per float4,
// global_prefetch_b8 one grid-stride ahead for extra MLP.

typedef float v4f __attribute__((ext_vector_type(4)));

__device__ __forceinline__ float fsu_abs_bit(float x, float c) {
    float t = __builtin_fmaf(2.0f, x, c - 1.0f);     // c + 2x - 1
    t = fminf(fmaxf(t, 0.0f), 15.0f);                // saturating counter [0, 15]
    return (t < 8.0f) ? (1.0f - x) : x;              // complement bit in negative region
}

__global__ __launch_bounds__(256)
void gaines_edge_kernel(const float* __restrict__ a,    // inp_Pr_i_j
                        const float* __restrict__ cnt,  // cnt_x
                        float* __restrict__ out,
                        long long n) {
    const long long n4     = n >> 2;
    const long long stride = (long long)gridDim.x * blockDim.x;
    long long i = (long long)blockIdx.x * blockDim.x + threadIdx.x;

    const v4f* __restrict__ a4 = (const v4f*)a;
    const v4f* __restrict__ c4 = (const v4f*)cnt;
    v4f*       __restrict__ o4 = (v4f*)out;

    for (; i < n4; i += stride) {
        // Prefetch next grid-stride iteration's cachelines (global_prefetch_b8).
        if (i + stride < n4) {
            __builtin_prefetch((const void*)(a4 + i + stride), 0, 0);
            __builtin_prefetch((const void*)(c4 + i + stride), 0, 0);
        }
        // Streaming (non-temporal) 128-bit loads: data is touched exactly once.
        v4f x = __builtin_nontemporal_load(a4 + i);
        v4f c = __builtin_nontemporal_load(c4 + i);
        v4f r;
#pragma unroll
        for (int k = 0; k < 4; ++k) r[k] = fsu_abs_bit(x[k], c[k]);
        __builtin_nontemporal_store(r, o4 + i);
    }

    // Scalar tail (n is a multiple of 4 for the reference shape; kept for generality).
    for (long long j = (n4 << 2) + (long long)blockIdx.x * blockDim.x + threadIdx.x;
         j < n; j += stride) {
        out[j] = fsu_abs_bit(a[j], cnt[j]);
    }
}

extern "C" void kernel_launch(void* const* d_in, const int* in_sizes, int n_in,
                              void* d_out, int out_size, void* d_ws, size_t ws_size,
                              hipStream_t stream) {
    (void)n_in; (void)d_ws; (void)ws_size;
    const float* a   = (const float*)d_in[0];   // inp_Pr_i_j
    const float* cnt = (const float*)d_in[4];   // cnt_x
    float*       out = (float*)d_out;
    const long long n = (long long)out_size;

    const int threads = 256;                     // 8 wave32 waves per block
    long long n4 = (n + 3) >> 2;
    long long want = (n4 + threads - 1) / threads;
    int blocks = (int)(want < 4096 ? (want > 0 ? want : 1) : 4096);

    gaines_edge_kernel<<<blocks, threads, 0, stream>>>(a, cnt, out, n);
}